// MultiCalibration_61993557950668
// MI455X (gfx1250) — compile-verified
//
#include <hip/hip_runtime.h>

typedef float v2f __attribute__((ext_vector_type(2)));
typedef float v8f __attribute__((ext_vector_type(8)));

namespace {
constexpr int kXDim   = 256;
constexpr int kTDim   = 4;
constexpr int kNFeats = 128;
constexpr int kB      = 2048;
constexpr int kCols   = kXDim * kNFeats;   // 32768
constexpr float kInvSqrtX = 0.0625f;       // 1/sqrt(256)
}

// One wave handles: 16 batch rows (M), one selector, 16 features (N).
// Loop over j (X_DIM) with V_WMMA_F32_16X16X4_F32 computing the 16x16 h-tile
// (K = T_DIM = 4) with bias folded in as the C matrix.
__global__ __launch_bounds__(32)
void fused_selector_wmma(const float* __restrict__ x,
                         const float* __restrict__ t,
                         const float* __restrict__ W1,
                         const float* __restrict__ b1,
                         const float* __restrict__ W2,
                         const float* __restrict__ b2,
                         const float* __restrict__ Wn1,
                         const float* __restrict__ Wn2,
                         float* __restrict__ mask_out,   // [B,2,256,128]
                         float* __restrict__ ypart)      // [B,2,8]
{
  __shared__ float xT[kXDim * 17];   // x tile transposed, pad 17 -> conflict-free

  const int lane = threadIdx.x;      // 0..31 (wave32)
  const int kg   = blockIdx.x;       // 0..7   feature group
  const int sel  = blockIdx.y;       // 0..1   selector
  const int it   = blockIdx.z;       // 0..127 batch tile
  const int i0   = it * 16;
  const int k0   = kg * 16;

  const float* __restrict__ W  = sel ? W2  : W1;
  const float* __restrict__ bs = sel ? b2  : b1;
  const float* __restrict__ Wn = sel ? Wn2 : Wn1;

  // Stage x[i0..i0+15, :] transposed into LDS: xT[j*17 + m] = x[i0+m, j]
  for (int idx = lane; idx < 16 * kXDim; idx += 32) {
    const int m = idx >> 8;            // 0..15
    const int j = idx & (kXDim - 1);   // 0..255
    xT[j * 17 + m] = x[(i0 + m) * kXDim + j];
  }
  __syncthreads();

  const int half = lane >> 4;   // 0: K rows 0,1 / 1: K rows 2,3 (A & B layout)
  const int ln   = lane & 15;

  // A matrix (16x4 f32): lane ln holds A[M=ln, K=2*half .. 2*half+1]
  v2f a;
  a.x = t[(i0 + ln) * kTDim + 2 * half + 0];
  a.y = t[(i0 + ln) * kTDim + 2 * half + 1];

  // ---- Pass 1: running max over j per (batch-row VGPR r, feature lane) ----
  float vmax[8];
#pragma unroll
  for (int r = 0; r < 8; ++r) vmax[r] = -__builtin_inff();

  for (int j = 0; j < kXDim; ++j) {
    const int col = j * kNFeats + k0 + ln;
    v2f bm;                                  // B (4x16): rows K striped by half
    bm.x = W[(2 * half + 0) * kCols + col];
    bm.y = W[(2 * half + 1) * kCols + col];
    const float bv = bs[col];
    v8f c;
#pragma unroll
    for (int r = 0; r < 8; ++r) c[r] = bv;   // bias broadcast across M
    v8f d = __builtin_amdgcn_wmma_f32_16x16x4_f32(false, a, false, bm,
                                                  (short)0, c, false, false);
#pragma unroll
    for (int r = 0; r < 8; ++r) vmax[r] = fmaxf(vmax[r], d[r]);
  }

  // ---- Pass 2: bit-identical recompute; write mask, gather x ----
  float yacc[8];
#pragma unroll
  for (int r = 0; r < 8; ++r) yacc[r] = 0.0f;

  for (int j = 0; j < kXDim; ++j) {
    const int col = j * kNFeats + k0 + ln;
    v2f bm;
    bm.x = W[(2 * half + 0) * kCols + col];
    bm.y = W[(2 * half + 1) * kCols + col];
    const float bv = bs[col];
    v8f c;
#pragma unroll
    for (int r = 0; r < 8; ++r) c[r] = bv;
    v8f d = __builtin_amdgcn_wmma_f32_16x16x4_f32(false, a, false, bm,
                                                  (short)0, c, false, false);
#pragma unroll
    for (int r = 0; r < 8; ++r) {
      const int   m  = r + half * 8;           // batch row within tile
      const float xv = xT[j * 17 + m];         // LDS broadcast per half-wave
      const bool  eq = (d[r] == vmax[r]);
      const float mv = eq ? 1.0f : 0.0f;
      const size_t midx =
          ((size_t)((i0 + m) * 2 + sel) * kXDim + j) * kNFeats + (k0 + ln);
      __builtin_nontemporal_store(mv, mask_out + midx);  // streaming 537MB
      yacc[r] += eq ? xv : 0.0f;
    }
  }

  // ---- Reduce: partial of y = sum_k (g[i,k] * Wn[k]) over this 16-feat group
  const float wk = Wn[k0 + ln];
#pragma unroll
  for (int r = 0; r < 8; ++r) {
    float p = yacc[r] * wk;
#pragma unroll
    for (int off = 1; off < 16; off <<= 1)
      p += __shfl_xor(p, off, 32);             // reduce within 16-lane half
    if (ln == 0) {
      const int i = i0 + r + half * 8;
      ypart[(i * 2 + sel) * 8 + kg] = p * kInvSqrtX;
    }
  }
}

// Deterministic fixed-order reduction of the 8 feature-group partials,
// plus the t-tail of the head and its bias.
__global__ __launch_bounds__(256)
void finalize_y(const float* __restrict__ t,
                const float* __restrict__ Wn1, const float* __restrict__ bn1,
                const float* __restrict__ Wn2, const float* __restrict__ bn2,
                const float* __restrict__ ypart,
                float* __restrict__ y_out)
{
  const int idx = blockIdx.x * blockDim.x + threadIdx.x;
  if (idx >= kB * 2) return;
  const int i = idx >> 1;
  const int s = idx & 1;
  const float* __restrict__ Wn = s ? Wn2 : Wn1;
  float acc = s ? bn2[0] : bn1[0];
#pragma unroll
  for (int d = 0; d < kTDim; ++d)
    acc += t[i * kTDim + d] * Wn[kNFeats + d];
#pragma unroll
  for (int kgi = 0; kgi < 8; ++kgi)
    acc += ypart[idx * 8 + kgi];
  y_out[idx] = acc;
}

extern "C" void kernel_launch(void* const* d_in, const int* in_sizes, int n_in,
                              void* d_out, int out_size, void* d_ws, size_t ws_size,
                              hipStream_t stream) {
  const float* x   = (const float*)d_in[0];
  const float* t   = (const float*)d_in[1];
  const float* W1  = (const float*)d_in[2];
  const float* b1  = (const float*)d_in[3];
  const float* W2  = (const float*)d_in[4];
  const float* b2  = (const float*)d_in[5];
  const float* Wn1 = (const float*)d_in[6];
  const float* bn1 = (const float*)d_in[7];
  const float* Wn2 = (const float*)d_in[8];
  const float* bn2 = (const float*)d_in[9];

  float* out      = (float*)d_out;
  float* mask_out = out + (size_t)kB * 2;      // y [B,2] first, then mask stack
  float* ypart    = (float*)d_ws;              // [B,2,8] partials (128 KB)

  dim3 grid(8, 2, kB / 16);                    // feat-group x selector x batch-tile
  fused_selector_wmma<<<grid, dim3(32, 1, 1), 0, stream>>>(
      x, t, W1, b1, W2, b2, Wn1, Wn2, mask_out, ypart);

  const int n = kB * 2;
  finalize_y<<<(n + 255) / 256, 256, 0, stream>>>(
      t, Wn1, bn1, Wn2, bn2, ypart, out);
}